// QuantumAttention_65481071404894
// MI455X (gfx1250) — compile-verified
//
#include <hip/hip_runtime.h>
#include <math.h>

typedef __attribute__((ext_vector_type(2))) float v2f;
typedef __attribute__((ext_vector_type(8))) float v8f;

#define B_  8
#define S_  2048
#define E_  8
#define H_  2
#define D_  4
#define NQ_ 8

// ---------------------------------------------------------------------------
// Kernel 1: fused q/k/v projections. q gets the 1/sqrt(D)=0.5 score scale
// folded in. Output layout [b][h][d][s] (s contiguous) so the attention
// kernel's WMMA operand loads are unit-stride along the sequence.
// ---------------------------------------------------------------------------
__global__ __launch_bounds__(256)
void proj_kernel(const float* __restrict__ x,
                 const float* __restrict__ Wq,
                 const float* __restrict__ Wk,
                 const float* __restrict__ Wv,
                 float* __restrict__ qw,
                 float* __restrict__ kw,
                 float* __restrict__ vw) {
  int t = blockIdx.x * blockDim.x + threadIdx.x;        // token id
  if (t >= B_ * S_) return;
  int b = t / S_, s = t % S_;
  float xv[E_];
#pragma unroll
  for (int e = 0; e < E_; ++e) xv[e] = x[(size_t)t * E_ + e];
#pragma unroll
  for (int f = 0; f < E_; ++f) {
    float q = 0.f, k = 0.f, v = 0.f;
#pragma unroll
    for (int e = 0; e < E_; ++e) {               // einsum 'bse,fe->bsf'
      q = fmaf(xv[e], Wq[f * E_ + e], q);
      k = fmaf(xv[e], Wk[f * E_ + e], k);
      v = fmaf(xv[e], Wv[f * E_ + e], v);
    }
    int h = f / D_, d = f % D_;
    size_t idx = ((((size_t)b * H_ + h) * D_ + d) * S_) + s;
    qw[idx] = q * 0.5f;                          // 1/sqrt(D) folded into q
    kw[idx] = k;
    vw[idx] = v;
  }
}

// ---------------------------------------------------------------------------
// Kernel 2: flash attention, fully transposed formulation. One wave32 per
// (b, h, 16-query block).
//
//   S^T = K_tile x Q^T        1x V_WMMA_F32_16X16X4_F32 (A=k 16x4, B=q^T 4x16)
//   O^T = V^T x P^T           4x V_WMMA_F32_16X16X4_F32 (K=4 chunks of 16 keys)
//
// In S^T's C/D layout each lane owns one query column i (= lane%16) and 8 key
// rows, so the online-softmax max is 7 v_max + ONE shfl_xor(16) per tile, and
// the running stats (m, alpha) are per-lane scalars.
//
// K-slot remap for the PV WMMAs: chunk c, lane-half H, slot v  ->  j=2c+8H+v.
// With that ordering the B operand of chunk c is {p[2c], p[2c+1]} in-register
// for every lane: no LDS, no shuffles for the P relayout.
//
// V^T A-operand carries a row of ones at d=4, so o[4] accumulates the softmax
// denominator through the same WMMAs (rescaled by alpha like the output).
// ---------------------------------------------------------------------------
__global__ __launch_bounds__(32)
void attn_kernel(const float* __restrict__ qw,
                 const float* __restrict__ kw,
                 const float* __restrict__ vw,
                 float* __restrict__ ow) {
  const int lane = threadIdx.x;
  const int half = lane >> 4;        // 0: lanes 0-15, 1: lanes 16-31
  const int m    = lane & 15;
  const int i0   = blockIdx.x * 16;
  const int h    = blockIdx.y;
  const int b    = blockIdx.z;

  const float* qbh = qw + (((size_t)b * H_ + h) * D_) * S_;
  const float* kbh = kw + (((size_t)b * H_ + h) * D_) * S_;
  const float* vbh = vw + (((size_t)b * H_ + h) * D_) * S_;

  // Q^T B-operand (K=d, N=i): loop invariant.  lane: N=m, K=2*half+{0,1}
  v2f qb;
  qb.x = qbh[(2 * half + 0) * S_ + i0 + m];
  qb.y = qbh[(2 * half + 1) * S_ + i0 + m];

  // V^T A-operand row select: rows 0-3 = V columns d, row 4 = ones (denom).
  const bool  mlt4   = (m < D_);
  const float onesel = (m == D_) ? 1.0f : 0.0f;
  const float* vrowp = vbh + (size_t)(m & (D_ - 1)) * S_;  // clamped d row

  v8f   o    = {};                   // O^T accumulator; o[4] = denominator
  float mrun = -3.402823466e+38f;    // per-lane running max for column i=m

  for (int j0 = 0; j0 < S_; j0 += 16) {
    // --- scores^T tile -----------------------------------------------------
    v2f ka;                          // A: K tile (M=j, K=d)
    ka.x = kbh[(2 * half + 0) * S_ + j0 + m];
    ka.y = kbh[(2 * half + 1) * S_ + j0 + m];
    v8f cz = {};
    v8f s = __builtin_amdgcn_wmma_f32_16x16x4_f32(
        false, ka, false, qb, (short)0, cz, false, false);

    // --- online softmax: per-lane stats ------------------------------------
    float mx = s[0];
#pragma unroll
    for (int r = 1; r < 8; ++r) mx = fmaxf(mx, s[r]);
    mx = fmaxf(mx, __shfl_xor(mx, 16));          // combine key halves
    float mnew = fmaxf(mrun, mx);
    float al   = __expf(mrun - mnew);
    mrun = mnew;

    float p[8];
#pragma unroll
    for (int r = 0; r < 8; ++r) p[r] = __expf(s[r] - mnew);

#pragma unroll
    for (int r = 0; r <= D_; ++r) o[r] *= al;    // rescale output + denom

    // --- O^T += V^T x P^T, K remap: chunk c, half H, slot v -> j=2c+8H+v ----
#pragma unroll
    for (int c = 0; c < 4; ++c) {
      int jj = j0 + 2 * c + 8 * half;
      float v0 = vrowp[jj + 0];
      float v1 = vrowp[jj + 1];
      v2f va, pb;
      va.x = mlt4 ? v0 : onesel;
      va.y = mlt4 ? v1 : onesel;
      pb.x = p[2 * c + 0];
      pb.y = p[2 * c + 1];
      o = __builtin_amdgcn_wmma_f32_16x16x4_f32(
          false, va, false, pb, (short)0, o, false, false);
    }
  }

  // --- normalize and store: lane m (< 16) owns token i0+m ------------------
  if (half == 0) {
    float inv = 1.0f / o[4];
    float4 r4 = make_float4(o[0] * inv, o[1] * inv, o[2] * inv, o[3] * inv);
    *(float4*)&ow[(((size_t)b * S_ + i0 + m) * E_) + h * D_] = r4;
  }
}

// ---------------------------------------------------------------------------
// Kernel 3: closed-form quantum measurement + Wo projection.
// The CNOT chain only permutes basis labels of a product state, so
//   <Z_q> = prod_{j in S_q} cos(out_j + out_{j mod 4}),
// with S_q = {0..q} for q>=1 and S_0 = {1..7}. Then out = z @ Wo^T.
// ---------------------------------------------------------------------------
__global__ __launch_bounds__(256)
void quantum_out_kernel(const float* __restrict__ ow,
                        const float* __restrict__ Wo,
                        float* __restrict__ out) {
  int t = blockIdx.x * blockDim.x + threadIdx.x;
  if (t >= B_ * S_) return;
  float o8[E_];
#pragma unroll
  for (int e = 0; e < E_; ++e) o8[e] = ow[(size_t)t * E_ + e];

  float c[NQ_];
#pragma unroll
  for (int j = 0; j < NQ_; ++j) c[j] = __cosf(o8[j] + o8[j & (D_ - 1)]);

  float z[NQ_];
  float pre = c[0];
#pragma unroll
  for (int q = 1; q < NQ_; ++q) { pre *= c[q]; z[q] = pre; }
  z[0] = c[1] * c[2] * c[3] * c[4] * c[5] * c[6] * c[7];

#pragma unroll
  for (int f = 0; f < E_; ++f) {    // einsum 'bsq,fq->bsf'
    float acc = 0.f;
#pragma unroll
    for (int q = 0; q < NQ_; ++q) acc = fmaf(z[q], Wo[f * NQ_ + q], acc);
    out[(size_t)t * E_ + f] = acc;
  }
}

// ---------------------------------------------------------------------------
extern "C" void kernel_launch(void* const* d_in, const int* in_sizes, int n_in,
                              void* d_out, int out_size, void* d_ws, size_t ws_size,
                              hipStream_t stream) {
  const float* x  = (const float*)d_in[0];
  const float* Wq = (const float*)d_in[1];
  const float* Wk = (const float*)d_in[2];
  const float* Wv = (const float*)d_in[3];
  const float* Wo = (const float*)d_in[4];
  float* out = (float*)d_out;

  const size_t NTOK = (size_t)B_ * S_;          // 16384
  const size_t PROJ = NTOK * E_;                // 131072 floats per tensor
  float* qw = (float*)d_ws;
  float* kw = qw + PROJ;
  float* vw = kw + PROJ;
  float* ow = vw + PROJ;                        // 4 * 512 KB = 2 MB total

  proj_kernel<<<dim3((NTOK + 255) / 256), dim3(256), 0, stream>>>(
      x, Wq, Wk, Wv, qw, kw, vw);

  attn_kernel<<<dim3(S_ / 16, H_, B_), dim3(32), 0, stream>>>(
      qw, kw, vw, ow);

  quantum_out_kernel<<<dim3((NTOK + 255) / 256), dim3(256), 0, stream>>>(
      ow, Wo, out);
}